// LCModel_25366076850607
// MI455X (gfx1250) — compile-verified
//
#include <hip/hip_runtime.h>
#include <stdint.h>

typedef unsigned int u32;

#define SORT_EPB     4096
#define SORT_THREADS 256
#define SCAN_CHUNK   4096
#define SCAN_THREADS 256
#define SCAN_RUN     16   // SCAN_CHUNK / SCAN_THREADS

// ---------------------------------------------------------------------------
// CDNA5 Tensor Data Mover: async global -> LDS staging (gfx1250 TDM path).
// 1-D tile descriptor per CDNA5 ISA (D# groups 0/1).
// This toolchain exposes the 6-arg builtin:
//   (uint32x4 g0, int32x8 g1, int32x4 g2, int32x4 g3, int32x8 g4, i32 cpol)
// ---------------------------------------------------------------------------
#if defined(__HIP_DEVICE_COMPILE__) && __has_builtin(__builtin_amdgcn_tensor_load_to_lds) && __has_builtin(__builtin_amdgcn_s_wait_tensorcnt)
#define USE_TDM 1
#else
#define USE_TDM 0
#endif

#if defined(__HIP_DEVICE_COMPILE__) && __has_builtin(__builtin_amdgcn_wmma_f32_16x16x4_f32)
#define USE_WMMA_RED 1
#else
#define USE_WMMA_RED 0
#endif

#if USE_TDM
typedef unsigned int u32x4 __attribute__((ext_vector_type(4)));
typedef int          i32x4 __attribute__((ext_vector_type(4)));
typedef int          i32x8 __attribute__((ext_vector_type(8)));

__device__ __forceinline__ void tdm_load_f32(void* lds_ptr, const float* gptr, u32 nelem) {
  u32 lds_off = (u32)(uintptr_t)lds_ptr;            // LDS offset = addr[31:0]
  uint64_t ga = (uint64_t)(uintptr_t)gptr;
  u32x4 g0;
  g0[0] = 1u;                                       // count=1, is_restore=0, gather=0
  g0[1] = lds_off;                                  // lds_addr
  g0[2] = (u32)ga;                                  // global_addr[31:0]
  g0[3] = (u32)((ga >> 32) & 0x01FFFFFFu) | (2u << 30); // global_addr[56:32] | type=2
  i32x8 g1;
  g1[0] = (int)(2u << 16);                          // wg_mask=0, data_size=4B
  g1[1] = (int)((nelem & 0xFFFFu) << 16);           // tensor_dim0[15:0]
  g1[2] = (int)(((nelem >> 16) & 0xFFFFu) | (1u << 16)); // tensor_dim0 hi | tensor_dim1=1
  g1[3] = (int)((nelem & 0xFFFFu) << 16);           // tile_dim0 = nelem (<=65535)
  g1[4] = 0;                                        // tile_dim1=0, tile_dim2=0
  g1[5] = (int)nelem;                               // tensor_dim0_stride lo
  g1[6] = 0;
  g1[7] = 0;
  i32x4 z4 = {0, 0, 0, 0};
  i32x8 z8 = {0, 0, 0, 0, 0, 0, 0, 0};
  __builtin_amdgcn_tensor_load_to_lds(g0, g1, z4, z4, z8, 0);
}
#endif

// ---------------------------------------------------------------------------
// small math helpers: 2x2 Mobius matrices (float4 = m00,m01,m10,m11) and
// affine pairs (float2 = a,b).  Combine is always newer * older.
// ---------------------------------------------------------------------------
__device__ __forceinline__ float4 mI() { return make_float4(1.f, 0.f, 0.f, 1.f); }
__device__ __forceinline__ float4 mmul(float4 B, float4 A) { // B newer
  return make_float4(B.x*A.x + B.y*A.z, B.x*A.y + B.y*A.w,
                     B.z*A.x + B.w*A.z, B.z*A.y + B.w*A.w);
}
__device__ __forceinline__ float4 mnorm(float4 M) { // projective normalization
  float s = fmaxf(fmaxf(fabsf(M.x), fabsf(M.y)), fmaxf(fabsf(M.z), fabsf(M.w)));
  float r = (s > 1e-30f) ? 1.f / s : 1.f;
  return make_float4(M.x*r, M.y*r, M.z*r, M.w*r);
}
__device__ __forceinline__ float2 aI() { return make_float2(1.f, 0.f); }
__device__ __forceinline__ float2 acomp(float2 nw, float2 od) { // newer o older
  return make_float2(nw.x * od.x, nw.x * od.y + nw.y);
}

// ===========================================================================
// Phase 1: sort keys (float -> order-preserving u32)
// ===========================================================================
__global__ void k_build_keys(const float* t, const int* band, const float* lag,
                             u32* keys, u32* vals, int N) {
  int n = blockIdx.x * blockDim.x + threadIdx.x;
  if (n >= N) return;
  int b = band[n];
  float lg = (b > 0) ? lag[b - 1] : 0.f;
  float nt = t[n] - lg;
  u32 u = __float_as_uint(nt);
  u ^= (u >> 31) ? 0xFFFFFFFFu : 0x80000000u;
  keys[n] = u;
  vals[n] = (u32)n;
}

__global__ void k_hist(const u32* keys, u32* hist, int shift, int nBlocks, int N) {
  __shared__ u32 h[256];
  int tid = threadIdx.x;
  h[tid] = 0;
  __syncthreads();
  int base = blockIdx.x * SORT_EPB;
  for (int i = base + tid; i < base + SORT_EPB; i += SORT_THREADS)
    if (i < N) atomicAdd(&h[(keys[i] >> shift) & 0xFFu], 1u);
  __syncthreads();
  hist[(size_t)tid * nBlocks + blockIdx.x] = h[tid];
}

// single-block exclusive scan over M u32s (M <= 1024*C)
__global__ void k_exscan(u32* data, int M) {
  __shared__ u32 arr[1024];
  int tid = threadIdx.x;
  int C = (M + 1023) >> 10;
  int s = tid * C;
  int e = s + C; if (e > M) e = M;
  u32 sum = 0;
  for (int i = s; i < e; i++) sum += data[i];
  arr[tid] = sum;
  __syncthreads();
  for (int off = 1; off < 1024; off <<= 1) {
    u32 v = arr[tid];
    u32 add = (tid >= off) ? arr[tid - off] : 0u;
    __syncthreads();
    arr[tid] = v + add;
    __syncthreads();
  }
  u32 run = arr[tid] - sum;
  for (int i = s; i < e; i++) { u32 v = data[i]; data[i] = run; run += v; }
}

// stable 8-bit scatter: local LSB bit-splits in LDS, then global placement
__global__ void k_scatter(const u32* keyIn, const u32* valIn, u32* keyOut, u32* valOut,
                          const u32* hist, int shift, int nBlocks, int N) {
  __shared__ u32 k0[SORT_EPB], v0[SORT_EPB], k1[SORT_EPB], v1[SORT_EPB];
  __shared__ u32 sarr[SORT_THREADS];
  __shared__ u32 dstart[256];
  __shared__ u32 gbase[256];
  int tid = threadIdx.x;
  int base = blockIdx.x * SORT_EPB;
  for (int i = tid; i < SORT_EPB; i += SORT_THREADS) {
    int g = base + i;
    k0[i] = (g < N) ? keyIn[g] : 0xFFFFFFFFu;
    v0[i] = (g < N) ? valIn[g] : 0xFFFFFFFFu;
  }
  __syncthreads();
  int cur = 0;
  for (int bit = 0; bit < 8; ++bit) {
    u32* ks = cur ? k1 : k0; u32* vs = cur ? v1 : v0;
    u32* kd = cur ? k0 : k1; u32* vd = cur ? v0 : v1;
    int lo = tid * 16;
    u32 cnt = 0, bits = 0;
    for (int j = 0; j < 16; ++j) {
      u32 f = (ks[lo + j] >> (shift + bit)) & 1u;
      bits |= f << j; cnt += f;
    }
    sarr[tid] = cnt;
    __syncthreads();
    for (int off = 1; off < SORT_THREADS; off <<= 1) {
      u32 v = sarr[tid];
      u32 add = (tid >= off) ? sarr[tid - off] : 0u;
      __syncthreads();
      sarr[tid] = v + add;
      __syncthreads();
    }
    u32 total = sarr[SORT_THREADS - 1];
    u32 run = sarr[tid] - cnt;        // ones before this thread's run
    u32 zbase = SORT_EPB - total;
    for (int j = 0; j < 16; ++j) {
      int l = lo + j;
      u32 f = (bits >> j) & 1u;
      u32 pos = f ? (zbase + run) : ((u32)l - run);
      kd[pos] = ks[l]; vd[pos] = vs[l];
      run += f;
    }
    __syncthreads();
    cur ^= 1;
  }
  // 8 flips -> sorted tile back in k0/v0
  for (int i = tid; i < SORT_EPB; i += SORT_THREADS) {
    u32 d = (k0[i] >> shift) & 0xFFu;
    if (i == 0 || (((k0[i - 1] >> shift) & 0xFFu) != d)) dstart[d] = (u32)i;
  }
  gbase[tid] = hist[(size_t)tid * nBlocks + blockIdx.x];
  __syncthreads();
  for (int i = tid; i < SORT_EPB; i += SORT_THREADS) {
    if (v0[i] == 0xFFFFFFFFu) continue;   // padding sentinel
    u32 d = (k0[i] >> shift) & 0xFFu;
    u32 pos = gbase[d] + ((u32)i - dstart[d]);
    keyOut[pos] = k0[i]; valOut[pos] = v0[i];
  }
}

// ===========================================================================
// Phase 2: gather into sorted order (contiguous arrays for the scans)
// ===========================================================================
__global__ void k_prep(const u32* idx, const float* t, const int* band, const float* diag,
                       const float* y, const float* lad, const float* lag,
                       float* ts_s, float* amp_s, float* diag_s, u32* y_s_bits, int N) {
  int n = blockIdx.x * blockDim.x + threadIdx.x;
  if (n >= N) return;
  u32 i = idx[n];                       // NOTE: y_s_bits aliases idx; read first
  int b = band[i];
  float la = (b > 0) ? lad[b - 1] : 0.f;
  float lg = (b > 0) ? lag[b - 1] : 0.f;
  float yv = y[i];
  float dv = diag[i];
  float tv = t[i] - lg;
  ts_s[n] = tv;
  amp_s[n] = expf(la);
  diag_s[n] = dv;
  y_s_bits[n] = __float_as_uint(yv);    // same-type store, safe aliasing
}

// ===========================================================================
// Phase 3: Mobius scan for S-state.
//   Sp' = ((A-2UV)p*Sp + V^2) / (-U^2 p*Sp + A),  p = phi^2
// ===========================================================================
#if USE_TDM
#define STAGE1(dst, src) tdm_load_f32(dst, (src) + base, SCAN_CHUNK)
#endif

__global__ void k_s1(const float* ts_s, const float* amp_s, const float* diag_s,
                     const float* lkp, float4* tmat, float4* bmat, int N) {
  __shared__ float lts[SCAN_CHUNK], lamp[SCAN_CHUNK], ldiag[SCAN_CHUNK];
  __shared__ float4 smat[SCAN_THREADS];
  int tid = threadIdx.x;
  int base = blockIdx.x * SCAN_CHUNK;
  float a = expf(lkp[0]), c = expf(lkp[1]);
#if USE_TDM
  if (tid == 0) {
    STAGE1(lts, ts_s); STAGE1(lamp, amp_s); STAGE1(ldiag, diag_s);
    __builtin_amdgcn_s_wait_tensorcnt(0);
  }
  __syncthreads();
#else
  for (int i = tid; i < SCAN_CHUNK; i += SCAN_THREADS) {
    int g = base + i;
    lts[i] = (g < N) ? ts_s[g] : 0.f;
    lamp[i] = (g < N) ? amp_s[g] : 0.f;
    ldiag[i] = (g < N) ? diag_s[g] : 1.f;
  }
  __syncthreads();
#endif
  int l0 = tid * SCAN_RUN, g0 = base + l0;
  float tprev = (g0 == 0) ? lts[0] : ((l0 == 0) ? ts_s[g0 - 1] : lts[l0 - 1]);
  float4 R = mI();
  for (int j = 0; j < SCAN_RUN; ++j) {
    int l = l0 + j, g = g0 + j;
    bool ok = (g < N);
    float amp = ok ? lamp[l] : 0.f;
    float Ad = ok ? (ldiag[l] + a * amp * amp) : 1.f;
    float U = a * amp, V = amp;
    float tsv = ok ? lts[l] : tprev;
    float phi = expf(-c * (tsv - tprev));
    tprev = tsv;
    float p = phi * phi;
    float4 M = make_float4((Ad - 2.f * U * V) * p, V * V, -(U * U) * p, Ad);
    R = mnorm(mmul(M, R));
  }
  smat[tid] = R;
  __syncthreads();
  for (int off = 1; off < SCAN_THREADS; off <<= 1) {
    float4 v = smat[tid];
    float4 comb = (tid >= off) ? mnorm(mmul(v, smat[tid - off])) : v;
    __syncthreads();
    smat[tid] = comb;
    __syncthreads();
  }
  tmat[(size_t)blockIdx.x * SCAN_THREADS + tid] = (tid > 0) ? smat[tid - 1] : mI();
  if (tid == SCAN_THREADS - 1) bmat[blockIdx.x] = smat[tid];
}

__global__ void k_mscan(const float4* bm, float4* bmx, int B) {
  __shared__ float4 arr[1024];
  int tid = threadIdx.x;
  arr[tid] = (tid < B) ? bm[tid] : mI();
  __syncthreads();
  for (int off = 1; off < 1024; off <<= 1) {
    float4 v = arr[tid];
    float4 comb = (tid >= off) ? mnorm(mmul(v, arr[tid - off])) : v;
    __syncthreads();
    arr[tid] = comb;
    __syncthreads();
  }
  if (tid < B) bmx[tid] = (tid > 0) ? arr[tid - 1] : mI();
}

// ===========================================================================
// Phase 4: affine scan for f-state.  f' = phi(1-WU) f + W r, W from Sp prefix.
// ===========================================================================
__global__ void k_f1(const float* ts_s, const float* amp_s, const float* diag_s,
                     const float* y_s, const float* lkp, const float4* tmat,
                     const float4* bmx, float2* taff, float2* baff, int N) {
  __shared__ float lts[SCAN_CHUNK], lamp[SCAN_CHUNK], ldiag[SCAN_CHUNK], ly[SCAN_CHUNK];
  __shared__ float2 saff[SCAN_THREADS];
  int tid = threadIdx.x;
  int base = blockIdx.x * SCAN_CHUNK;
  float a = expf(lkp[0]), c = expf(lkp[1]);
#if USE_TDM
  if (tid == 0) {
    STAGE1(lts, ts_s); STAGE1(lamp, amp_s); STAGE1(ldiag, diag_s); STAGE1(ly, y_s);
    __builtin_amdgcn_s_wait_tensorcnt(0);
  }
  __syncthreads();
#else
  for (int i = tid; i < SCAN_CHUNK; i += SCAN_THREADS) {
    int g = base + i;
    lts[i] = (g < N) ? ts_s[g] : 0.f;
    lamp[i] = (g < N) ? amp_s[g] : 0.f;
    ldiag[i] = (g < N) ? diag_s[g] : 1.f;
    ly[i] = (g < N) ? y_s[g] : 0.f;
  }
  __syncthreads();
#endif
  float4 E = mmul(tmat[(size_t)blockIdx.x * SCAN_THREADS + tid], bmx[blockIdx.x]);
  float x = E.y / E.w;                               // Sp entering this run
  int l0 = tid * SCAN_RUN, g0 = base + l0;
  float tprev = (g0 == 0) ? lts[0] : ((l0 == 0) ? ts_s[g0 - 1] : lts[l0 - 1]);
  float2 Rf = aI();
  for (int j = 0; j < SCAN_RUN; ++j) {
    int l = l0 + j, g = g0 + j;
    bool ok = (g < N);
    float amp = ok ? lamp[l] : 0.f;
    float Ad = ok ? (ldiag[l] + a * amp * amp) : 1.f;
    float U = a * amp, V = amp;
    float tsv = ok ? lts[l] : tprev;
    float r = ok ? ly[l] : 0.f;
    float phi = expf(-c * (tsv - tprev));
    tprev = tsv;
    float p = phi * phi;
    float S = p * x;
    float D = Ad - U * U * S;
    float W = (V - U * S) / D;
    Rf = acomp(make_float2(phi * (1.f - W * U), W * r), Rf);
    x = S + D * W * W;
  }
  saff[tid] = Rf;
  __syncthreads();
  for (int off = 1; off < SCAN_THREADS; off <<= 1) {
    float2 v = saff[tid];
    float2 comb = (tid >= off) ? acomp(v, saff[tid - off]) : v;
    __syncthreads();
    saff[tid] = comb;
    __syncthreads();
  }
  taff[(size_t)blockIdx.x * SCAN_THREADS + tid] = (tid > 0) ? saff[tid - 1] : aI();
  if (tid == SCAN_THREADS - 1) baff[blockIdx.x] = saff[tid];
}

__global__ void k_ascan(const float2* ba, float2* bax, int B) {
  __shared__ float2 arr[1024];
  int tid = threadIdx.x;
  arr[tid] = (tid < B) ? ba[tid] : aI();
  __syncthreads();
  for (int off = 1; off < 1024; off <<= 1) {
    float2 v = arr[tid];
    float2 comb = (tid >= off) ? acomp(v, arr[tid - off]) : v;
    __syncthreads();
    arr[tid] = comb;
    __syncthreads();
  }
  if (tid < B) bax[tid] = (tid > 0) ? arr[tid - 1] : aI();
}

// ===========================================================================
// Phase 5: final pass — z, D, per-block partial sums (deterministic tree)
// ===========================================================================
__global__ void k_f3(const float* ts_s, const float* amp_s, const float* diag_s,
                     const float* y_s, const float* lkp, const float4* tmat,
                     const float4* bmx, const float2* taff, const float2* bax,
                     float2* part, int N) {
  __shared__ float lts[SCAN_CHUNK], lamp[SCAN_CHUNK], ldiag[SCAN_CHUNK], ly[SCAN_CHUNK];
  __shared__ float2 red[SCAN_THREADS];
  int tid = threadIdx.x;
  int base = blockIdx.x * SCAN_CHUNK;
  float a = expf(lkp[0]), c = expf(lkp[1]);
#if USE_TDM
  if (tid == 0) {
    STAGE1(lts, ts_s); STAGE1(lamp, amp_s); STAGE1(ldiag, diag_s); STAGE1(ly, y_s);
    __builtin_amdgcn_s_wait_tensorcnt(0);
  }
  __syncthreads();
#else
  for (int i = tid; i < SCAN_CHUNK; i += SCAN_THREADS) {
    int g = base + i;
    lts[i] = (g < N) ? ts_s[g] : 0.f;
    lamp[i] = (g < N) ? amp_s[g] : 0.f;
    ldiag[i] = (g < N) ? diag_s[g] : 1.f;
    ly[i] = (g < N) ? y_s[g] : 0.f;
  }
  __syncthreads();
#endif
  float4 E = mmul(tmat[(size_t)blockIdx.x * SCAN_THREADS + tid], bmx[blockIdx.x]);
  float x = E.y / E.w;
  float2 tot = acomp(taff[(size_t)blockIdx.x * SCAN_THREADS + tid], bax[blockIdx.x]);
  float f = tot.y;                                   // f entering run (f0 = 0)
  int l0 = tid * SCAN_RUN, g0 = base + l0;
  float tprev = (g0 == 0) ? lts[0] : ((l0 == 0) ? ts_s[g0 - 1] : lts[l0 - 1]);
  float s1 = 0.f, s2 = 0.f;
  for (int j = 0; j < SCAN_RUN; ++j) {
    int l = l0 + j, g = g0 + j;
    bool ok = (g < N);
    float amp = ok ? lamp[l] : 0.f;
    float Ad = ok ? (ldiag[l] + a * amp * amp) : 1.f;
    float U = a * amp, V = amp;
    float tsv = ok ? lts[l] : tprev;
    float r = ok ? ly[l] : 0.f;
    float phi = expf(-c * (tsv - tprev));
    tprev = tsv;
    float p = phi * phi;
    float S = p * x;
    float D = Ad - U * U * S;
    float W = (V - U * S) / D;
    float fv = phi * f;
    float z = r - U * fv;
    s1 += z * z / D;
    s2 += logf(D);
    f = fv + W * z;
    x = S + D * W * W;
  }
  red[tid] = make_float2(s1, s2);
  __syncthreads();
  for (int off = SCAN_THREADS / 2; off > 0; off >>= 1) {
    if (tid < off) { red[tid].x += red[tid + off].x; red[tid].y += red[tid + off].y; }
    __syncthreads();
  }
  if (tid == 0) part[blockIdx.x] = red[0];
}

// ===========================================================================
// Final reduction over per-block partials.  One full wave (32 lanes).
// WMMA path: C += A(ones-weighted data) x B(ones) with V_WMMA_F32_16X16X4_F32.
//   s1 partials -> rows 0..7, s2 partials -> rows 8..15; row sums land in C,
//   identical across columns; lanes 0-15 hold rows 0-7, lanes 16-31 rows 8-15.
//   Fully deterministic, full f32 precision.
// ===========================================================================
#if USE_WMMA_RED
typedef float v2f __attribute__((ext_vector_type(2)));
typedef float v8f __attribute__((ext_vector_type(8)));
#endif

__global__ void k_reduce(const float2* part, int B, float* out, float n) {
  __shared__ float sh[32];
  int lane = threadIdx.x;            // blockDim.x == 32, one wave, EXEC all 1s
#if USE_WMMA_RED
  int M = lane & 15;
  int kbase = (lane < 16) ? 0 : 2;   // A layout: lanes 0-15 K={0,1}, 16-31 K={2,3}
  v2f bones; bones[0] = 1.f; bones[1] = 1.f;
  v8f acc = {0.f, 0.f, 0.f, 0.f, 0.f, 0.f, 0.f, 0.f};
  int iters = (B + 31) / 32;
  for (int it = 0; it < iters; ++it) {
    int base = it * 32;
    v2f av;
    for (int r = 0; r < 2; ++r) {
      int k = kbase + r;
      float v;
      if (M < 8) {                   // rows 0..7 carry s1 partials
        int slot = base + M * 4 + k;
        v = (slot < B) ? part[slot].x : 0.f;
      } else {                       // rows 8..15 carry s2 partials
        int slot = base + (M - 8) * 4 + k;
        v = (slot < B) ? part[slot].y : 0.f;
      }
      av[r] = v;
    }
    acc = __builtin_amdgcn_wmma_f32_16x16x4_f32(
        /*neg_a=*/false, av, /*neg_b=*/false, bones,
        /*c_mod=*/(short)0, acc, /*reuse_a=*/false, /*reuse_b=*/false);
  }
  // lane<16: acc[r] = rowsum(row r)   -> sum = total s1
  // lane>=16: acc[r] = rowsum(row 8+r)-> sum = total s2
  float s = acc[0] + acc[1] + acc[2] + acc[3] + acc[4] + acc[5] + acc[6] + acc[7];
  sh[lane] = s;
  __syncthreads();
  if (lane == 0)
    out[0] = 0.5f * (sh[0] + sh[16] + n * 1.8378770664093454f); // log(2*pi)
#else
  float s1 = 0.f, s2 = 0.f;
  for (int i = lane; i < B; i += 32) { s1 += part[i].x; s2 += part[i].y; }
  sh[lane] = s1;
  __syncthreads();
  for (int off = 16; off > 0; off >>= 1) {
    if (lane < off) sh[lane] += sh[lane + off];
    __syncthreads();
  }
  float t1 = sh[0];
  __syncthreads();
  sh[lane] = s2;
  __syncthreads();
  for (int off = 16; off > 0; off >>= 1) {
    if (lane < off) sh[lane] += sh[lane + off];
    __syncthreads();
  }
  if (lane == 0)
    out[0] = 0.5f * (t1 + sh[0] + n * 1.8378770664093454f); // log(2*pi)
#endif
}

// ===========================================================================
// Host launcher
// ===========================================================================
extern "C" void kernel_launch(void* const* d_in, const int* in_sizes, int n_in,
                              void* d_out, int out_size, void* d_ws, size_t ws_size,
                              hipStream_t stream) {
  const float* t    = (const float*)d_in[0];
  const int*   band = (const int*)d_in[1];
  const float* y    = (const float*)d_in[2];
  const float* diag = (const float*)d_in[3];
  const float* lad  = (const float*)d_in[4];
  const float* lag  = (const float*)d_in[5];
  const float* lkp  = (const float*)d_in[6];
  float* out = (float*)d_out;
  const int N = in_sizes[0];
  const int nSB = (N + SORT_EPB - 1) / SORT_EPB;    // 1024 for N=4M
  const int nCB = (N + SCAN_CHUNK - 1) / SCAN_CHUNK; // 1024 for N=4M
  (void)n_in; (void)out_size; (void)ws_size;

  char* w = (char*)d_ws;
  auto al = [](size_t x) { return (x + 255) & ~(size_t)255; };
  size_t o = 0;
  size_t szN4 = al((size_t)N * 4);
  u32* kA = (u32*)(w + o); o += szN4;
  u32* vA = (u32*)(w + o); o += szN4;
  u32* kB = (u32*)(w + o); o += szN4;
  u32* vB = (u32*)(w + o); o += szN4;
  u32* hist = (u32*)(w + o); o += al((size_t)256 * nSB * 4);
  float4* tmat = (float4*)(w + o); o += al((size_t)nCB * SCAN_THREADS * 16);
  float4* bmat = (float4*)(w + o); o += al((size_t)nCB * 16);
  float4* bmx  = (float4*)(w + o); o += al((size_t)nCB * 16);
  float2* taff = (float2*)(w + o); o += al((size_t)nCB * SCAN_THREADS * 8);
  float2* baff = (float2*)(w + o); o += al((size_t)nCB * 8);
  float2* bax  = (float2*)(w + o); o += al((size_t)nCB * 8);
  float2* part = (float2*)(w + o); o += al((size_t)nCB * 8);

  // 1) keys
  k_build_keys<<<(N + 255) / 256, 256, 0, stream>>>(t, band, lag, kA, vA, N);

  // 2) 4-pass LSD radix sort (stable) — sorted index ends in vA
  u32 *ks = kA, *vs = vA, *kd = kB, *vd = vB;
  for (int pass = 0; pass < 4; ++pass) {
    int shift = pass * 8;
    k_hist<<<nSB, SORT_THREADS, 0, stream>>>(ks, hist, shift, nSB, N);
    k_exscan<<<1, 1024, 0, stream>>>(hist, 256 * nSB);
    k_scatter<<<nSB, SORT_THREADS, 0, stream>>>(ks, vs, kd, vd, hist, shift, nSB, N);
    u32* tk = ks; ks = kd; kd = tk;
    u32* tv = vs; vs = vd; vd = tv;
  }
  // ks==kA, vs==vA now; keys dead -> recycle buffers for sorted arrays
  float* ts_s   = (float*)ks;   // kA
  float* amp_s  = (float*)kd;   // kB
  float* diag_s = (float*)vd;   // vB
  float* y_s    = (float*)vs;   // vA (aliases idx; prep reads idx first)

  // 3) gather into sorted order
  k_prep<<<(N + 255) / 256, 256, 0, stream>>>((const u32*)vs, t, band, diag, y,
                                              lad, lag, ts_s, amp_s, diag_s,
                                              (u32*)y_s, N);

  // 4) Mobius prefix scan (S-state), 3 levels
  k_s1<<<nCB, SCAN_THREADS, 0, stream>>>(ts_s, amp_s, diag_s, lkp, tmat, bmat, N);
  k_mscan<<<1, 1024, 0, stream>>>(bmat, bmx, nCB);

  // 5) affine prefix scan (f-state), 3 levels
  k_f1<<<nCB, SCAN_THREADS, 0, stream>>>(ts_s, amp_s, diag_s, y_s, lkp,
                                         tmat, bmx, taff, baff, N);
  k_ascan<<<1, 1024, 0, stream>>>(baff, bax, nCB);

  // 6) final pass: z, log D, partial sums; then WMMA reduction (one wave)
  k_f3<<<nCB, SCAN_THREADS, 0, stream>>>(ts_s, amp_s, diag_s, y_s, lkp,
                                         tmat, bmx, taff, bax, part, N);
  k_reduce<<<1, 32, 0, stream>>>(part, nCB, out, (float)N);
}